// RNNLayer_25744033973053
// MI455X (gfx1250) — compile-verified
//
#include <hip/hip_runtime.h>
#include <math.h>

typedef __attribute__((ext_vector_type(2))) float v2f;
typedef __attribute__((ext_vector_type(8))) float v8f;
typedef __attribute__((ext_vector_type(4))) int   v4i;

#define BB 64
#define SS 512
#define DD 768
#define HH 768
#define KTOT 1536            // D + H concatenated
#define BK 64                // K-chunk
#define NCHUNK (KTOT / BK)   // 24 (chunks 0..11 from v, 12..23 from h)
#define ASTRIDE 68           // LDS row stride: conflict-free (68*4 % 64 banks) and 16B aligned

#if defined(__AMDGCN__) && __has_builtin(__builtin_amdgcn_global_load_async_to_lds_b128)
#define HAS_ASYNC_LDS 1
#else
#define HAS_ASYNC_LDS 0
#endif

__device__ __forceinline__ float sigmoidf_fast(float x) {
    return 1.0f / (1.0f + __expf(-x));
}

// Zero h ping-pong buffers and cell state.
__global__ void lstm_init_kernel(float* h0, float* h1, float* c) {
    int i = blockIdx.x * blockDim.x + threadIdx.x;
    if (i < BB * HH) { h0[i] = 0.0f; h1[i] = 0.0f; c[i] = 0.0f; }
}

// Prefetch one 16x64 A-chunk of concat [v_t | h_in] into LDS buffer `buf`.
// 128 threads x 2 float4 = 16 rows x 64 cols. Async (ASYNCcnt) when available.
__device__ __forceinline__ void issue_chunk(int kc, float (*Ald)[16][ASTRIDE], int buf,
                                            int tid, int m0, int t,
                                            const float* __restrict__ v,
                                            const float* __restrict__ h_in) {
#pragma unroll
    for (int half = 0; half < 2; ++half) {
        const int fi  = tid + half * 128;
        const int row = fi >> 4;
        const int col = (fi & 15) * 4;
        const int k   = kc * BK + col;
        const float* src = (k < DD)
            ? (v + ((size_t)(m0 + row) * SS + t) * DD + k)
            : (h_in + (size_t)(m0 + row) * HH + (k - DD));
        float* dst = &Ald[buf][row][col];
#if HAS_ASYNC_LDS
        __builtin_amdgcn_global_load_async_to_lds_b128(
            (v4i*)const_cast<float*>(src), (v4i*)dst, 0, 0);
#else
        *(float4*)dst = *(const float4*)src;
#endif
    }
}

#if HAS_ASYNC_LDS
// Each chunk = 2 async ops per thread; in steady state only the just-issued
// prefetch (2 ops) may remain outstanding.
#define WAIT_ASYNC_PREV() asm volatile("s_wait_asynccnt 0x2" ::: "memory")
#define WAIT_ASYNC_ALL()  asm volatile("s_wait_asynccnt 0x0" ::: "memory")
#else
#define WAIT_ASYNC_PREV()
#define WAIT_ASYNC_ALL()
#endif

// 16 K=4 WMMA steps over one 16x64 chunk. wpLane is this lane's weight-row
// pointer (already offset by 2*lh) at the chunk's K base -> the 16 float2
// loads become immediate-offset loads off one base.
__device__ __forceinline__ void chunk_mma(const float (*A)[ASTRIDE],
                                          const float* __restrict__ wpLane,
                                          v8f& acc, int lm, int lh) {
#pragma unroll
    for (int kk = 0; kk < 16; ++kk) {
        v2f a;
        a.x = A[lm][kk * 4 + 2 * lh];
        a.y = A[lm][kk * 4 + 2 * lh + 1];
        const float2 bw = *(const float2*)(wpLane + kk * 4);
        v2f b;
        b.x = bw.x;
        b.y = bw.y;
        acc = __builtin_amdgcn_wmma_f32_16x16x4_f32(
            false, a, false, b, (short)0, acc, false, false);
    }
}

// One LSTM timestep, fused: gates = [v_t ; h_in] @ [W_ih ; W_hh]^T + b,
// then pointwise i,f,g,o update with length-mask semantics.
// grid = (48 n-tiles, 4 m-tiles), block = 128 threads = 4 waves (one per gate).
__global__ void __launch_bounds__(128)
lstm_step_kernel(const float* __restrict__ v, const int* __restrict__ mask,
                 const float* __restrict__ W_ih, const float* __restrict__ W_hh,
                 const float* __restrict__ b_ih, const float* __restrict__ b_hh,
                 const float* __restrict__ h_in, float* __restrict__ h_out,
                 float* __restrict__ c_ws, float* __restrict__ out, int t) {
    __shared__ float Ald[2][16][ASTRIDE];   // double-buffered 16x64 A tile
    __shared__ float gld[4][16][17];        // per-gate 16x16 result tiles

    const int tid  = threadIdx.x;
    const int wave = tid >> 5;              // gate index: 0=i,1=f,2=g,3=o
    const int lane = tid & 31;
    const int lm = lane & 15;
    const int lh = lane >> 4;

    const int n0 = blockIdx.x * 16;         // hidden-column tile
    const int m0 = blockIdx.y * 16;         // batch tile

    // Bias depends only on the output column; init accumulator with it.
    const int wrow = wave * HH + n0 + lm;   // W row for this lane's column
    const float bias = b_ih[wrow] + b_hh[wrow];
    v8f acc;
#pragma unroll
    for (int r = 0; r < 8; ++r) acc[r] = bias;

    // Per-lane weight-row base pointers (K-major rows, L2-resident), pre-offset
    // by this lane's K phase (lanes 0-15 -> K{0,1}, lanes 16-31 -> K{2,3}).
    const float* wiLane = W_ih + (size_t)wrow * DD + 2 * lh;
    const float* whLane = W_hh + (size_t)wrow * HH + 2 * lh;

    issue_chunk(0, Ald, 0, tid, m0, t, v, h_in);

#pragma unroll 1
    for (int kc = 0; kc < NCHUNK; ++kc) {
        const int buf = kc & 1;
        if (kc + 1 < NCHUNK) {
            issue_chunk(kc + 1, Ald, buf ^ 1, tid, m0, t, v, h_in);
            WAIT_ASYNC_PREV();
        } else {
            WAIT_ASYNC_ALL();
        }
        __syncthreads();
        // Uniform per-chunk select: v-half (chunks 0..11) vs h-half (12..23).
        const float* wp = (kc < DD / BK) ? (wiLane + kc * BK)
                                         : (whLane + (kc - DD / BK) * BK);
        chunk_mma(Ald[buf], wp, acc, lm, lh);
        __syncthreads();
    }

    // Spill this wave's gate tile to LDS: C/D layout row m = r + 8*lh, col n = lm
#pragma unroll
    for (int r = 0; r < 8; ++r) gld[wave][r + 8 * lh][lm] = acc[r];
    __syncthreads();

    // Pointwise LSTM update: 256 elements, 2 per thread
    for (int e = tid; e < 256; e += 128) {
        const int m = e >> 4;
        const int n = e & 15;
        const int bi  = m0 + m;             // batch index
        const int col = n0 + n;             // hidden index
        const float ig = sigmoidf_fast(gld[0][m][n]);
        const float fg = sigmoidf_fast(gld[1][m][n]);
        const float gg = tanhf(gld[2][m][n]);
        const float og = sigmoidf_fast(gld[3][m][n]);
        const size_t hidx = (size_t)bi * HH + col;
        const float c_old = c_ws[hidx];
        const float h_old = h_in[hidx];
        const float c_new = fg * c_old + ig * gg;
        const float h_new = og * tanhf(c_new);
        const int mt = mask[(size_t)bi * SS + t];
        h_out[hidx] = mt ? h_new : h_old;
        c_ws[hidx]  = mt ? c_new : c_old;
        out[((size_t)bi * SS + t) * HH + col] = mt ? h_new : 0.0f;
    }
}

// o_last = final hidden state (state freezes past each sequence length)
__global__ void lstm_final_kernel(const float* __restrict__ h, float* __restrict__ o_last) {
    int i = blockIdx.x * blockDim.x + threadIdx.x;
    if (i < BB * HH) o_last[i] = h[i];
}

extern "C" void kernel_launch(void* const* d_in, const int* in_sizes, int n_in,
                              void* d_out, int out_size, void* d_ws, size_t ws_size,
                              hipStream_t stream) {
    const float* v    = (const float*)d_in[0];
    const int*   mask = (const int*)d_in[1];
    const float* W_ih = (const float*)d_in[2];
    const float* W_hh = (const float*)d_in[3];
    const float* b_ih = (const float*)d_in[4];
    const float* b_hh = (const float*)d_in[5];

    float* h0 = (float*)d_ws;            // ping
    float* h1 = h0 + (size_t)BB * HH;    // pong
    float* cs = h1 + (size_t)BB * HH;    // cell state

    float* out    = (float*)d_out;                         // (B,S,H)
    float* o_last = out + (size_t)BB * SS * HH;            // (B,H)

    lstm_init_kernel<<<(BB * HH + 255) / 256, 256, 0, stream>>>(h0, h1, cs);

    dim3 grid(HH / 16, BB / 16);   // 48 x 4
    for (int t = 0; t < SS; ++t) {
        const float* hin = (t & 1) ? h1 : h0;
        float*       hou = (t & 1) ? h0 : h1;
        lstm_step_kernel<<<grid, 128, 0, stream>>>(v, mask, W_ih, W_hh, b_ih, b_hh,
                                                   hin, hou, cs, out, t);
    }
    // S=512 even: last step (t=511) read h1, wrote h0 -> final h is h0
    lstm_final_kernel<<<(BB * HH + 255) / 256, 256, 0, stream>>>(h0, o_last);
}